// GraphClassifierSelfDistillation_42528766165148
// MI455X (gfx1250) — compile-verified
//
#include <hip/hip_runtime.h>
#include <hip/hip_bf16.h>

#define NN 50000
#define EE 800000
#define DD 128
#define LL 3
#define GG 512
#define CC 10
#define BN_EPS 1e-5f

typedef float v2f __attribute__((ext_vector_type(2)));
typedef float v8f __attribute__((ext_vector_type(8)));

// ---------------- utility ----------------
__global__ void zero_kernel(float* __restrict__ p, long n) {
    long i = (long)blockIdx.x * blockDim.x + threadIdx.x;
    if (i < n) p[i] = 0.0f;
}

__global__ void deg_kernel(const int* __restrict__ dst, float* __restrict__ deg) {
    int e = blockIdx.x * blockDim.x + threadIdx.x;
    if (e < EE) atomicAdd(&deg[dst[e]], 1.0f);
}

__global__ void gcnt_kernel(const int* __restrict__ batch, float* __restrict__ gcnt) {
    int i = blockIdx.x * blockDim.x + threadIdx.x;
    if (i < NN) atomicAdd(&gcnt[batch[i]], 1.0f);
}

// one block per edge; 128 threads = one feature row
__global__ __launch_bounds__(128) void scatter_kernel(const float* __restrict__ h,
                                                      const int* __restrict__ src,
                                                      const int* __restrict__ dst,
                                                      float* __restrict__ agg) {
    int e = blockIdx.x;
    int s = src[e], d = dst[e];
    atomicAdd(&agg[(size_t)d * DD + threadIdx.x], h[(size_t)s * DD + threadIdx.x]);
}

// ---------------- fused SAGE GEMM: hnew = (agg/deg)@Wl + bl + h@Wr ----------------
// one wave per 16-row tile, full 128 output columns, fp32 WMMA 16x16x4
__global__ __launch_bounds__(32) void sage_gemm_kernel(const float* __restrict__ agg,
                                                       const float* __restrict__ h,
                                                       const float* __restrict__ deg,
                                                       const float* __restrict__ Wl,
                                                       const float* __restrict__ Wr,
                                                       const float* __restrict__ bl,
                                                       float* __restrict__ hnew) {
    const int lane = threadIdx.x;        // 0..31
    const int m    = lane & 15;          // row/col within tile
    const int koff = (lane >> 4) << 1;   // 0 for lanes 0-15, 2 for lanes 16-31
    const int row0 = blockIdx.x * 16;
    const int row  = row0 + m;

    const float invd = 1.0f / fmaxf(deg[row], 1.0f);
    const float* aggRow = agg + (size_t)row * DD;
    const float* hRow   = h   + (size_t)row * DD;

    v8f acc[8];
#pragma unroll
    for (int ct = 0; ct < 8; ++ct) {
#pragma unroll
        for (int r = 0; r < 8; ++r) acc[ct][r] = 0.0f;
    }

    for (int kb = 0; kb < DD; kb += 4) {
        const int k0 = kb + koff;
        v2f a1, a2;
        a1.x = aggRow[k0] * invd;  a1.y = aggRow[k0 + 1] * invd;  // neighbor mean
        a2.x = hRow[k0];           a2.y = hRow[k0 + 1];           // root
#pragma unroll
        for (int ct = 0; ct < 8; ++ct) {
            const int col = ct * 16 + m;
            v2f b1, b2;
            b1.x = Wl[(size_t)k0 * DD + col];
            b1.y = Wl[(size_t)(k0 + 1) * DD + col];
            b2.x = Wr[(size_t)k0 * DD + col];
            b2.y = Wr[(size_t)(k0 + 1) * DD + col];
            acc[ct] = __builtin_amdgcn_wmma_f32_16x16x4_f32(false, a1, false, b1,
                                                            (short)0, acc[ct], false, false);
            acc[ct] = __builtin_amdgcn_wmma_f32_16x16x4_f32(false, a2, false, b2,
                                                            (short)0, acc[ct], false, false);
        }
    }

    const int rbase = (lane >> 4) << 3;  // C/D layout: upper lane half holds M = r+8
#pragma unroll
    for (int ct = 0; ct < 8; ++ct) {
        const int col  = ct * 16 + m;
        const float bias = bl[col];
#pragma unroll
        for (int r = 0; r < 8; ++r) {
            hnew[(size_t)(row0 + rbase + r) * DD + col] = acc[ct][r] + bias;
        }
    }
}

// ---------------- BatchNorm column statistics ----------------
__global__ __launch_bounds__(128) void bnstat_kernel(const float* __restrict__ hnew,
                                                     float* __restrict__ colsum,
                                                     float* __restrict__ colsq) {
    int c = threadIdx.x;
    float s = 0.0f, s2 = 0.0f;
    for (int r = blockIdx.x; r < NN; r += gridDim.x) {
        float v = hnew[(size_t)r * DD + c];
        s += v; s2 += v * v;
    }
    atomicAdd(&colsum[c], s);
    atomicAdd(&colsq[c],  s2);
}

// ---------------- normalize + ReLU + graph pooling (accumulate) ----------------
__global__ __launch_bounds__(128) void bnorm_pool_kernel(const float* __restrict__ hnew,
                                                         const float* __restrict__ colsum,
                                                         const float* __restrict__ colsq,
                                                         const float* __restrict__ gamma,
                                                         const float* __restrict__ beta,
                                                         const int* __restrict__ batch,
                                                         float* __restrict__ hout,
                                                         float* __restrict__ feat) {
    int c = threadIdx.x;
    const float invN = 1.0f / (float)NN;
    float mean = colsum[c] * invN;
    float var  = colsq[c] * invN - mean * mean;
    float rs   = rsqrtf(var + BN_EPS);
    float gm = gamma[c], bt = beta[c];
    for (int r = blockIdx.x; r < NN; r += gridDim.x) {
        float v = hnew[(size_t)r * DD + c];
        v = gm * (v - mean) * rs + bt;
        v = fmaxf(v, 0.0f);
        hout[(size_t)r * DD + c] = v;
        atomicAdd(&feat[(size_t)batch[r] * DD + c], v);
    }
}

__global__ void featscale_kernel(float* __restrict__ feats, const float* __restrict__ gcnt) {
    int i = blockIdx.x * blockDim.x + threadIdx.x;   // over L*G*D
    if (i < LL * GG * DD) {
        int g = (i / DD) % GG;
        feats[i] /= fmaxf(gcnt[g], 1.0f);
    }
}

// ---------------- per-depth classifier head: 128 -> 64 -> 32 -> 10 ----------------
__global__ __launch_bounds__(64) void head_kernel(const float* __restrict__ feats,
                                                  const float* __restrict__ w1, const float* __restrict__ b1,
                                                  const float* __restrict__ w2, const float* __restrict__ b2,
                                                  const float* __restrict__ w3, const float* __restrict__ b3,
                                                  float* __restrict__ outs) {
    __shared__ float sf[DD];
    __shared__ float sf1[64];
    __shared__ float sf2[32];

    int l = blockIdx.x / GG;
    int g = blockIdx.x % GG;
    int t = threadIdx.x;

    const float* frow = feats + ((size_t)l * GG + g) * DD;
    sf[t]      = frow[t];
    sf[t + 64] = frow[t + 64];
    __syncthreads();

    // fc1: 128 -> 64
    {
        float a = b1[l * 64 + t];
        const float* W = w1 + (size_t)l * DD * 64;
        for (int k = 0; k < DD; ++k) a += sf[k] * W[(size_t)k * 64 + t];
        sf1[t] = fmaxf(a, 0.0f);
    }
    __syncthreads();

    // fc2: 64 -> 32
    if (t < 32) {
        float a = b2[l * 32 + t];
        const float* W = w2 + (size_t)l * 64 * 32;
        for (int k = 0; k < 64; ++k) a += sf1[k] * W[(size_t)k * 32 + t];
        sf2[t] = fmaxf(a, 0.0f);
    }
    __syncthreads();

    // fc3: 32 -> 10
    if (t < CC) {
        float a = b3[l * CC + t];
        const float* W = w3 + (size_t)l * 32 * CC;
        for (int k = 0; k < 32; ++k) a += sf2[k] * W[(size_t)k * CC + t];
        outs[((size_t)l * GG + g) * CC + t] = a;
    }
}

// ---------------- driver ----------------
extern "C" void kernel_launch(void* const* d_in, const int* in_sizes, int n_in,
                              void* d_out, int out_size, void* d_ws, size_t ws_size,
                              hipStream_t stream) {
    const float* x     = (const float*)d_in[0];
    const int*   ei    = (const int*)d_in[1];      // [2,E]: row0 = src, row1 = dst
    const int*   batch = (const int*)d_in[2];
    const float* Wl    = (const float*)d_in[3];
    const float* bl    = (const float*)d_in[4];
    const float* Wr    = (const float*)d_in[5];
    const float* gamma = (const float*)d_in[6];
    const float* beta  = (const float*)d_in[7];
    const float* w1    = (const float*)d_in[8];
    const float* b1    = (const float*)d_in[9];
    const float* w2    = (const float*)d_in[10];
    const float* b2    = (const float*)d_in[11];
    const float* w3    = (const float*)d_in[12];
    const float* b3    = (const float*)d_in[13];

    const int* src = ei;
    const int* dst = ei + EE;

    float* outs  = (float*)d_out;                  // [L,G,C]
    float* feats = outs + (size_t)LL * GG * CC;    // [L,G,D]

    // workspace carve-up
    float* agg    = (float*)d_ws;                  // N*D
    float* hnew   = agg  + (size_t)NN * DD;        // N*D
    float* hbuf   = hnew + (size_t)NN * DD;        // N*D  (normalized h for next layer)
    float* deg    = hbuf + (size_t)NN * DD;        // N
    float* gcnt   = deg  + NN;                     // G
    float* colsum = gcnt + GG;                     // D
    float* colsq  = colsum + DD;                   // D

    const long nd = (long)NN * DD;

    // graph-structure quantities + zero accumulation targets
    zero_kernel<<<(NN + 255) / 256, 256, 0, stream>>>(deg, NN);
    zero_kernel<<<(GG + 255) / 256, 256, 0, stream>>>(gcnt, GG);
    zero_kernel<<<(int)((LL * GG * DD + 255) / 256), 256, 0, stream>>>(feats, (long)LL * GG * DD);
    deg_kernel<<<(EE + 255) / 256, 256, 0, stream>>>(dst, deg);
    gcnt_kernel<<<(NN + 255) / 256, 256, 0, stream>>>(batch, gcnt);

    const float* hcur = x;
    for (int i = 0; i < LL; ++i) {
        zero_kernel<<<(int)((nd + 255) / 256), 256, 0, stream>>>(agg, nd);
        scatter_kernel<<<EE, 128, 0, stream>>>(hcur, src, dst, agg);

        sage_gemm_kernel<<<NN / 16, 32, 0, stream>>>(agg, hcur, deg,
                                                     Wl + (size_t)i * DD * DD,
                                                     Wr + (size_t)i * DD * DD,
                                                     bl + (size_t)i * DD,
                                                     hnew);

        zero_kernel<<<1, 256, 0, stream>>>(colsum, 2 * DD);   // colsum + colsq contiguous
        bnstat_kernel<<<512, 128, 0, stream>>>(hnew, colsum, colsq);

        bnorm_pool_kernel<<<1024, 128, 0, stream>>>(hnew, colsum, colsq,
                                                    gamma + (size_t)i * DD,
                                                    beta + (size_t)i * DD,
                                                    batch, hbuf,
                                                    feats + (size_t)i * GG * DD);
        hcur = hbuf;
    }

    featscale_kernel<<<(LL * GG * DD + 255) / 256, 256, 0, stream>>>(feats, gcnt);
    head_kernel<<<LL * GG, 64, 0, stream>>>(feats, w1, b1, w2, b2, w3, b3, outs);
}